// UnsupervisedMorphemeSegmenter_13297218748817
// MI455X (gfx1250) — compile-verified
//
#include <hip/hip_runtime.h>
#include <hip/hip_bf16.h>
#include <stddef.h>

// Problem constants (match reference)
#define BB 512
#define LL 512
#define HH 256
#define MM 24
#define NEGV (-1.0e9f)

#define AS1 __attribute__((address_space(1)))
#define AS3 __attribute__((address_space(3)))

typedef float v2f __attribute__((ext_vector_type(2)));
typedef float v8f __attribute__((ext_vector_type(8)));
typedef int   v4i __attribute__((ext_vector_type(4)));

// ---------------------------------------------------------------------------
// Kernel 1: scores[b,l] = dot(we[b,l,:], W) + bias, masked to NEG for
// l >= wl[b]-1.  One wave (32 lanes) per row; each lane loads 8 contiguous
// floats (two float4), shuffle-reduce across the wave32.
// ---------------------------------------------------------------------------
__global__ void __launch_bounds__(256) scores_kernel(
    const float* __restrict__ we, const int* __restrict__ wlens,
    const float* __restrict__ W, const float* __restrict__ bias,
    float* __restrict__ scores)
{
    const int wave = threadIdx.x >> 5;
    const int lane = threadIdx.x & 31;
    const int row  = blockIdx.x * 8 + wave;          // 0 .. B*L-1

    const float4* p = (const float4*)(we + (size_t)row * HH + lane * 8);
    const float4* q = (const float4*)(W + lane * 8);
    float4 x0 = p[0], x1 = p[1];
    float4 w0 = q[0], w1 = q[1];
    float acc = x0.x * w0.x + x0.y * w0.y + x0.z * w0.z + x0.w * w0.w
              + x1.x * w1.x + x1.y * w1.y + x1.z * w1.z + x1.w * w1.w;
    for (int off = 16; off > 0; off >>= 1)
        acc += __shfl_xor(acc, off, 32);

    if (lane == 0) {
        int b = row >> 9;            // / LL
        int l = row & (LL - 1);
        float s = acc + bias[0];
        if (l >= wlens[b] - 1) s = NEGV;
        scores[row] = s;
    }
}

// ---------------------------------------------------------------------------
// Kernel 2: per batch row: iterative top-23 argmax (smallest-index tiebreak,
// matching jax.lax.top_k), boundary markers at s+1, inclusive cumsum ->
// char2morph, emit bpm (0/1 floats) and sentinel-masked c2m for kernel 3.
// ---------------------------------------------------------------------------
__global__ void __launch_bounds__(256) topk_bpm_kernel(
    const float* __restrict__ scores, const int* __restrict__ wlens,
    int* __restrict__ c2m_g, float* __restrict__ bpm_out)
{
    __shared__ float sval[LL];
    __shared__ int   bnd[LL];
    __shared__ int   c2ms[LL];
    __shared__ float rv[256];
    __shared__ int   ri[256];
    __shared__ int   sc[2][256];

    const int b = blockIdx.x;
    const int t = threadIdx.x;

    sval[t]       = scores[b * LL + t];
    sval[t + 256] = scores[b * LL + 256 + t];
    bnd[t] = 0;
    bnd[t + 256] = 0;
    __syncthreads();

    // 23 sequential argmax selections
    for (int it = 0; it < MM - 1; ++it) {
        float v0 = sval[t], v1 = sval[t + 256];
        float v; int i;
        if (v1 > v0) { v = v1; i = t + 256; } else { v = v0; i = t; }
        rv[t] = v; ri[t] = i;
        __syncthreads();
        for (int off = 128; off > 0; off >>= 1) {
            if (t < off) {
                float ov = rv[t + off]; int oi = ri[t + off];
                if (ov > rv[t] || (ov == rv[t] && oi < ri[t])) {
                    rv[t] = ov; ri[t] = oi;
                }
            }
            __syncthreads();
        }
        if (t == 0) {
            int s = ri[0];
            if (s + 1 < LL) bnd[s + 1] = 1;   // boundary at char s+1
            sval[s] = -3.0e38f;               // remove from next selection
        }
        __syncthreads();
    }

    // inclusive cumsum of bnd -> char2morph (pair-sum + Hillis-Steele scan)
    int e0 = bnd[2 * t], e1 = bnd[2 * t + 1];
    sc[0][t] = e0 + e1;
    __syncthreads();
    int src = 0;
    for (int off = 1; off < 256; off <<= 1) {
        int v = sc[src][t];
        if (t >= off) v += sc[src][t - off];
        sc[1 - src][t] = v;
        src = 1 - src;
        __syncthreads();
    }
    int incl = sc[src][t];                    // cumsum through element 2t+1
    c2ms[2 * t + 1] = incl;
    c2ms[2 * t]     = incl - e1;
    __syncthreads();

    const int wl = wlens[b];
    c2m_g[b * LL + t]       = (t       < wl) ? c2ms[t]       : 255;
    c2m_g[b * LL + 256 + t] = (t + 256 < wl) ? c2ms[t + 256] : 255;

    float* out = bpm_out + (size_t)b * LL * MM;
    for (int idx = t; idx < LL * MM; idx += 256) {
        int l = idx / MM;
        int m = idx - l * MM;
        out[idx] = (l < wl && c2ms[l] == m) ? 1.0f : 0.0f;
    }
}

// ---------------------------------------------------------------------------
// Kernel 3: menc[b] = bpm_b^T (24x512, padded to 32x512) @ we_b (512x256)
// using V_WMMA_F32_16X16X4_F32 (exact f32 matrix path).  One block per batch.
// we_b is streamed HBM -> LDS with GLOBAL_LOAD_ASYNC_TO_LDS_B128 (ASYNCcnt),
// double-buffered in 16-row k-chunks so chunk k+1 is in flight while WMMA
// consumes chunk k.  LDS row stride 272 keeps the two half-wave B-fragment
// reads on disjoint bank sets.  8 waves x 4 tiles cover 2 (M) x 16 (N) tiles.
// ---------------------------------------------------------------------------
#define KCH 16
#define LSTR 272
#define NCH (LL / KCH)

__global__ void __launch_bounds__(256) menc_wmma_kernel(
    const float* __restrict__ we, const int* __restrict__ c2m_g,
    float* __restrict__ menc)
{
    __shared__ float swe[2][KCH * LSTR];
    __shared__ int   sc2m[LL];

    const int b    = blockIdx.x;
    const int t    = threadIdx.x;
    const int wave = t >> 5;
    const int lane = t & 31;
    const int half = lane >> 4;      // 0: lanes 0-15, 1: lanes 16-31
    const int col  = lane & 15;

    sc2m[t]       = c2m_g[b * LL + t];
    sc2m[t + 256] = c2m_g[b * LL + 256 + t];

    v8f zero8 = {0.f, 0.f, 0.f, 0.f, 0.f, 0.f, 0.f, 0.f};
    v8f acc[4] = {zero8, zero8, zero8, zero8};

    const float* web = we + (size_t)b * LL * HH;

    // async HBM -> LDS: one 16-row x 256-col chunk = 1024 b128 transfers,
    // 4 per thread (16B each, 16B-aligned on both sides)
    auto issue_chunk = [&](int ck, int bf) {
        #pragma unroll
        for (int q = 0; q < 4; ++q) {
            int id  = q * 256 + t;       // 0..1023
            int row = id >> 6;           // 64 x b128 per 256-float row
            int cg  = id & 63;
            __builtin_amdgcn_global_load_async_to_lds_b128(
                (AS1 v4i*)(web + (size_t)(ck * KCH + row) * HH + cg * 4),
                (AS3 v4i*)(&swe[bf][row * LSTR + cg * 4]),
                0, 0);
        }
    };

    issue_chunk(0, 0);                   // prefetch chunk 0

    for (int ci = 0; ci < NCH; ++ci) {
        const int buf = ci & 1;
        const int kc  = ci * KCH;
        __syncthreads();                 // everyone done with chunk ci-1 (buf^1 free)
        if (ci + 1 < NCH) {
            issue_chunk(ci + 1, buf ^ 1);
            // 4 newest in-flight loads are chunk ci+1; in-order completion
            // means <=4 outstanding implies chunk ci is resident.
            asm volatile("s_wait_asynccnt 4" ::: "memory");
        } else {
            asm volatile("s_wait_asynccnt 0" ::: "memory");
        }
        __syncthreads();                 // chunk ci visible to all waves

        #pragma unroll
        for (int j = 0; j < 4; ++j) {
            const int tid = wave + 8 * j;      // 0..31
            const int mt  = tid >> 4;          // 0..1  (M tile)
            const int nt  = tid & 15;          // 0..15 (N tile)
            const int m   = mt * 16 + col;     // this lane's A row
            #pragma unroll
            for (int ks = 0; ks < KCH; ks += 4) {
                const int l0 = ks + 2 * half;  // K split across half-waves
                v2f a, bf;
                a.x  = (sc2m[kc + l0]     == m) ? 1.0f : 0.0f;
                a.y  = (sc2m[kc + l0 + 1] == m) ? 1.0f : 0.0f;
                bf.x = swe[buf][l0       * LSTR + nt * 16 + col];
                bf.y = swe[buf][(l0 + 1) * LSTR + nt * 16 + col];
                acc[j] = __builtin_amdgcn_wmma_f32_16x16x4_f32(
                    false, a, false, bf, (short)0, acc[j], false, false);
            }
        }
    }

    // write D tiles: VGPR r -> rows r (lanes 0-15) / r+8 (lanes 16-31)
    #pragma unroll
    for (int j = 0; j < 4; ++j) {
        const int tid = wave + 8 * j;
        const int mt  = tid >> 4;
        const int nt  = tid & 15;
        #pragma unroll
        for (int r = 0; r < 8; ++r) {
            int mrow = mt * 16 + r + 8 * half;
            if (mrow < MM) {
                menc[((size_t)b * MM + mrow) * HH + nt * 16 + col] = acc[j][r];
            }
        }
    }
}

// ---------------------------------------------------------------------------
extern "C" void kernel_launch(void* const* d_in, const int* in_sizes, int n_in,
                              void* d_out, int out_size, void* d_ws, size_t ws_size,
                              hipStream_t stream) {
    (void)in_sizes; (void)n_in; (void)out_size; (void)ws_size;

    const float* we    = (const float*)d_in[0];   // [B, L, H] f32
    const int*   wlens = (const int*)d_in[1];     // [B] i32
    // d_in[2] = num_morphemes (constant == MM, unused)
    const float* W     = (const float*)d_in[3];   // [H, 1] f32
    const float* bias  = (const float*)d_in[4];   // [1] f32

    float* menc = (float*)d_out;                  // [B*M, H] f32
    float* bpm  = menc + (size_t)BB * MM * HH;    // [B, L, M] as 0/1 f32

    float* scores = (float*)d_ws;                              // B*L f32
    int*   c2m    = (int*)((char*)d_ws + (size_t)BB * LL * 4); // B*L i32

    scores_kernel<<<(BB * LL) / 8, 256, 0, stream>>>(we, wlens, W, bias, scores);
    topk_bpm_kernel<<<BB, 256, 0, stream>>>(scores, wlens, c2m, bpm);
    menc_wmma_kernel<<<BB, 256, 0, stream>>>(we, c2m, menc);
}